// FixedComplexCell_18167711662682
// MI455X (gfx1250) — compile-verified
//
#include <hip/hip_runtime.h>

// Elementwise complex gate: pure streaming kernel for MI455X (gfx1250).
// Roofline: 24 B/elem * 16.78M elems ~= 403 MB -> ~17.3 us floor @ 23.3 TB/s;
// ~12 flops/elem -> far below any compute ceiling. No matmul structure ->
// WMMA/TDM inapplicable. Optimal CDNA5 shape: wave32 + b128 NT loads/stores,
// 4 float4s per thread to amortize the per-block scalar preamble and keep
// 16 independent loads in flight per wave.

typedef float v4f __attribute__((ext_vector_type(4)));

#define GATE_EPS 1e-8f
#define TPB     256
#define UNROLL  4

__global__ __launch_bounds__(TPB)
void complex_gate_kernel(const float* __restrict__ z_re,
                         const float* __restrict__ z_im,
                         const float* __restrict__ gate,
                         float* __restrict__ out,
                         int D, long long N)
{
    const int row  = blockIdx.y;
    const int col0 = blockIdx.x * (TPB * 4 * UNROLL) + threadIdx.x * 4;

    // Hoisted per-row 64-bit bases; inner offsets become immediate offsets.
    const size_t zrow = (size_t)row * (size_t)D;
    const size_t grow = (size_t)row * 2u * (size_t)D;

    const float* __restrict__ pzr = z_re + zrow + col0;
    const float* __restrict__ pzm = z_im + zrow + col0;
    const float* __restrict__ pg0 = gate + grow + col0;
    const float* __restrict__ pg1 = pg0 + (size_t)D;
    float* __restrict__ pore = out + zrow + col0;
    float* __restrict__ poim = pore + (size_t)N;

#pragma unroll
    for (int u = 0; u < UNROLL; ++u) {
        const int col = col0 + u * (TPB * 4);
        if (col < D) {
            const int off = u * (TPB * 4);

            // Streamed once, never reused: non-temporal b128 loads.
            const v4f zr = __builtin_nontemporal_load((const v4f*)(pzr + off));
            const v4f zm = __builtin_nontemporal_load((const v4f*)(pzm + off));
            const v4f g0 = __builtin_nontemporal_load((const v4f*)(pg0 + off));
            const v4f g1 = __builtin_nontemporal_load((const v4f*)(pg1 + off));

            v4f ore, oim;
#pragma unroll
            for (int k = 0; k < 4; ++k) {
                const float a = g0[k];
                const float b = g1[k];
                // inv = 1/sqrt(a^2 + b^2 + eps)  -> v_rsq_f32 path
                const float inv = rsqrtf(fmaf(a, a, fmaf(b, b, GATE_EPS)));
                const float gr = a * inv;
                const float gm = b * inv;
                // (zr + i*zm) * (gr + i*gm)
                ore[k] = fmaf(zr[k], gr, -(zm[k] * gm));
                oim[k] = fmaf(zr[k], gm,   zm[k] * gr);
            }

            // Outputs never re-read: non-temporal b128 stores.
            __builtin_nontemporal_store(ore, (v4f*)(pore + off));
            __builtin_nontemporal_store(oim, (v4f*)(poim + off));
        }
    }
}

extern "C" void kernel_launch(void* const* d_in, const int* in_sizes, int n_in,
                              void* d_out, int out_size, void* d_ws, size_t ws_size,
                              hipStream_t stream)
{
    const float* z_re = (const float*)d_in[0];
    const float* z_im = (const float*)d_in[1];
    const float* gate = (const float*)d_in[2];
    float*       out  = (float*)d_out;

    const int       D = 4096;                   // reference: (4096, 4096)
    const long long N = (long long)in_sizes[0]; // B*D elements per plane
    const int       B = (int)(N / D);

    const int vec_per_row   = D / 4;                         // 1024
    const int vec_per_block = TPB * UNROLL;                  // 1024
    dim3 block(TPB, 1, 1);
    dim3 grid((unsigned)((vec_per_row + vec_per_block - 1) / vec_per_block),
              (unsigned)B, 1);

    hipLaunchKernelGGL(complex_gate_kernel, grid, block, 0, stream,
                       z_re, z_im, gate, out, D, N);
}